// DeformConv3D_19791209299991
// MI455X (gfx1250) — compile-verified
//
#include <hip/hip_runtime.h>
#include <cmath>

// Problem constants (from reference): B=4, C=64, D=8, H=W=64, G=4, KS=3
#define G      4
#define KS     3
#define K      9        // KS*KS
#define CG     16       // channels per group
#define CPG    18       // offset-conv channels per group = 2*K
#define H      64
#define W      64
#define HW     4096
#define NSLICE 32       // B*D
#define COUT   64
#define EPS    1e-5f

#define TILE   64       // pixels per workgroup in stage 2
#define CKTOT  (CG*K)   // 144 reduction length
#define WSTR   32       // LDS row stride (dwords) for weight ck-pairs: 16 o's * 2
#define VSTR   160      // LDS row stride (dwords) for val ck-pairs: 64 pix * 2 + 32 pad (bank-conflict free)

typedef __attribute__((ext_vector_type(2))) float v2f;
typedef __attribute__((ext_vector_type(8))) float v8f;

// ---------------------------------------------------------------------------
// Stage 1: grouped 3x3 offset conv.  off[n][g*18+ko][hw] = conv(x) + b_off
// ---------------------------------------------------------------------------
__global__ __launch_bounds__(256) void offset_conv_kernel(
    const float* __restrict__ x, const float* __restrict__ w_off,
    const float* __restrict__ b_off, float* __restrict__ off)
{
    __shared__ float wl[CG * K];                 // 144 weights for this output row
    const int bx  = blockIdx.x;                  // n*72 + row
    const int n   = bx / (G * CPG);
    const int row = bx % (G * CPG);              // g*18 + ko
    const int g   = row / CPG;
    if (threadIdx.x < CG * K) wl[threadIdx.x] = w_off[row * (CG * K) + threadIdx.x];
    __syncthreads();

    const int hw = blockIdx.y * 256 + threadIdx.x;
    const int h  = hw >> 6, w = hw & 63;
    const float* xb = x + (size_t)(n * COUT + g * CG) * HW;
    float acc = b_off[row];
    #pragma unroll
    for (int i = 0; i < KS; ++i) {
        int y = h - 1 + i;
        if ((unsigned)y >= (unsigned)H) continue;
        #pragma unroll
        for (int j = 0; j < KS; ++j) {
            int xx = w - 1 + j;
            if ((unsigned)xx >= (unsigned)W) continue;
            const float* xp = xb + y * W + xx;
            #pragma unroll
            for (int c = 0; c < CG; ++c)
                acc = fmaf(wl[c * K + i * KS + j], xp[c * HW], acc);
        }
    }
    off[(size_t)(n * (G * CPG) + row) * HW + hw] = acc;
}

// ---------------------------------------------------------------------------
// Stage 2: bilinear sampling into LDS + fp32 WMMA GEMM
//   out[16o, 64pix] = W[16o, 144ck] * val[144ck, 64pix]     (per n, g, tile)
// Block: 128 threads = 4 waves; wave w computes pixels [16w, 16w+16).
// LDS layouts store K-pairs contiguously so fragments load as 8B LDS reads:
//   wl[(ck>>1)*WSTR + o*2 + (ck&1)]     (lanes16-31 offset by 32 dwords -> banks 32..63)
//   vl[(ck>>1)*VSTR + p*2 + (ck&1)]     (VSTR=160 -> +2 rows shifts banks by 32)
// WMMA f32 16x16x4 lane layout (ISA 7.12.2): lanes0-15 hold K=+0/+1 (m=lane),
// lanes16-31 hold K=+2/+3 (m=lane-16); D VGPR i = rows M=i (lo lanes) / M=i+8.
// ---------------------------------------------------------------------------
__global__ __launch_bounds__(128) void deform_wmma_kernel(
    const float* __restrict__ x, const float* __restrict__ off,
    const float* __restrict__ w_dc, const float* __restrict__ b_dc,
    float* __restrict__ mid)
{
    __shared__ float wl[(CKTOT / 2) * WSTR];     // 72*32  dwords =  9.2 KB
    __shared__ float vl[(CKTOT / 2) * VSTR];     // 72*160 dwords = 46.1 KB

    const int tile = blockIdx.x & 63;
    const int g    = (blockIdx.x >> 6) & 3;
    const int n    = blockIdx.x >> 8;
    const int t    = threadIdx.x;

    // ---- load W_dc[g] (16 x 144) into pair-interleaved LDS ----
    for (int idx = t; idx < CG * CKTOT; idx += 128) {
        int o  = idx / CKTOT;
        int ck = idx % CKTOT;
        wl[(ck >> 1) * WSTR + o * 2 + (ck & 1)] = w_dc[(g * CG + o) * CKTOT + ck];
    }

    // ---- bilinear sampling: val[ck = c*9+k][p] for 64 pixels ----
    const int hw0 = tile * TILE;
    const float* xb   = x   + (size_t)(n * COUT + g * CG) * HW;
    const float* offb = off + (size_t)(n * (G * CPG) + g * CPG) * HW;
    for (int pair = t; pair < TILE * K; pair += 128) {   // 576 (pixel,tap) pairs
        int p  = pair / K;
        int k  = pair % K;
        int hw = hw0 + p;
        int h  = hw >> 6, w = hw & 63;
        int ki = k / 3,  kj = k % 3;
        float dy = offb[(2 * k + 0) * HW + hw];
        float dx = offb[(2 * k + 1) * HW + hw];
        float py = (float)(h - 1 + ki) + dy;
        float px = (float)(w - 1 + kj) + dx;
        float fy = floorf(py), fx = floorf(px);
        float ly = py - fy,    lx = px - fx;
        int y0 = (int)fy, x0 = (int)fx;
        int y1 = y0 + 1,  x1 = x0 + 1;
        float vy0 = ((unsigned)y0 < (unsigned)H) ? 1.f : 0.f;
        float vy1 = ((unsigned)y1 < (unsigned)H) ? 1.f : 0.f;
        float vx0 = ((unsigned)x0 < (unsigned)W) ? 1.f : 0.f;
        float vx1 = ((unsigned)x1 < (unsigned)W) ? 1.f : 0.f;
        float w00 = (1.f - ly) * (1.f - lx) * vy0 * vx0;
        float w01 = (1.f - ly) * lx         * vy0 * vx1;
        float w10 = ly * (1.f - lx)         * vy1 * vx0;
        float w11 = ly * lx                 * vy1 * vx1;
        int cy0 = min(max(y0, 0), H - 1), cy1 = min(max(y1, 0), H - 1);
        int cx0 = min(max(x0, 0), W - 1), cx1 = min(max(x1, 0), W - 1);
        int i00 = cy0 * W + cx0, i01 = cy0 * W + cx1;
        int i10 = cy1 * W + cx0, i11 = cy1 * W + cx1;
        #pragma unroll
        for (int c = 0; c < CG; ++c) {
            const float* xp = xb + c * HW;
            float v = w00 * xp[i00] + w01 * xp[i01] + w10 * xp[i10] + w11 * xp[i11];
            int ck = c * K + k;
            vl[(ck >> 1) * VSTR + p * 2 + (ck & 1)] = v;
        }
    }
    __syncthreads();

    // ---- WMMA GEMM: 36 x v_wmma_f32_16x16x4_f32 per wave ----
    const int lane  = t & 31;
    const int wave  = t >> 5;
    const int khalf = lane >> 4;   // 0: K+0/+1 lanes, 1: K+2/+3 lanes
    const int mn    = lane & 15;   // A row (o) / B col (pixel)
    const int pbase = wave * 16;

    v8f acc = {0.f, 0.f, 0.f, 0.f, 0.f, 0.f, 0.f, 0.f};
    #pragma unroll 4
    for (int ck = 0; ck < CKTOT; ck += 4) {
        int ckp = (ck + 2 * khalf) >> 1;   // even pair index for this lane half
        v2f a = *(const v2f*)&wl[ckp * WSTR + mn * 2];
        v2f b = *(const v2f*)&vl[ckp * VSTR + (pbase + mn) * 2];
        acc = __builtin_amdgcn_wmma_f32_16x16x4_f32(
                  /*neg_a=*/false, a, /*neg_b=*/false, b,
                  /*c_mod=*/(short)0, acc, /*reuse_a=*/false, /*reuse_b=*/false);
    }

    // ---- store D + bias: VGPR i -> rows o=i (lanes<16) / o=i+8 (lanes>=16) ----
    const int hw = hw0 + pbase + mn;
    #pragma unroll
    for (int i = 0; i < 8; ++i) {
        int o  = i + 8 * khalf;
        int oc = g * CG + o;
        mid[(size_t)(n * COUT + oc) * HW + hw] = acc[i] + b_dc[oc];
    }
}

// ---------------------------------------------------------------------------
// Stage 3: per-(n,c) spatial mean/var normalization + exact GELU + transpose
// out[b][c][d][h][w] = gelu((mid[n=b*8+d][c][hw] - mean) * rsqrt(var + eps))
// ---------------------------------------------------------------------------
__global__ __launch_bounds__(256) void norm_gelu_kernel(
    const float* __restrict__ mid, float* __restrict__ out)
{
    __shared__ float s1[256], s2[256];
    const int nc = blockIdx.x;               // n*64 + c
    const int n  = nc >> 6, c = nc & 63;
    const int t  = threadIdx.x;
    const float* p = mid + (size_t)nc * HW;

    float vals[16];
    float sum = 0.f, sq = 0.f;
    #pragma unroll
    for (int i = 0; i < 16; ++i) {
        float v = p[t + i * 256];
        vals[i] = v;
        sum += v;
        sq  += v * v;
    }
    s1[t] = sum; s2[t] = sq;
    __syncthreads();
    for (int o = 128; o > 0; o >>= 1) {
        if (t < o) { s1[t] += s1[t + o]; s2[t] += s2[t + o]; }
        __syncthreads();
    }
    const float mean = s1[0] * (1.f / HW);
    const float var  = s2[0] * (1.f / HW) - mean * mean;
    const float rstd = rsqrtf(var + EPS);

    const int b = n >> 3, d = n & 7;
    float* op = out + ((size_t)(b * COUT + c) * 8 + d) * HW;
    #pragma unroll
    for (int i = 0; i < 16; ++i) {
        float z = (vals[i] - mean) * rstd;
        op[t + i * 256] = 0.5f * z * (1.f + erff(z * 0.70710678118654752f));
    }
}

// ---------------------------------------------------------------------------
extern "C" void kernel_launch(void* const* d_in, const int* in_sizes, int n_in,
                              void* d_out, int out_size, void* d_ws, size_t ws_size,
                              hipStream_t stream) {
    (void)in_sizes; (void)n_in; (void)out_size; (void)ws_size;
    const float* x     = (const float*)d_in[0];
    const float* w_off = (const float*)d_in[1];
    const float* b_off = (const float*)d_in[2];
    const float* w_dc  = (const float*)d_in[3];
    const float* b_dc  = (const float*)d_in[4];
    float* out = (float*)d_out;

    // Workspace: off (32*72*4096 fl = 36 MB) then mid (32*64*4096 fl = 32 MB)
    float* off = (float*)d_ws;
    float* mid = off + (size_t)NSLICE * (G * CPG) * HW;

    dim3 g1(NSLICE * G * CPG, HW / 256);          // (2304, 16)
    offset_conv_kernel<<<g1, 256, 0, stream>>>(x, w_off, b_off, off);

    deform_wmma_kernel<<<NSLICE * G * (HW / TILE), 128, 0, stream>>>(
        x, off, w_dc, b_dc, mid);

    norm_gelu_kernel<<<NSLICE * COUT, 256, 0, stream>>>(mid, out);
}